// ChannelPatchEmbedding_25692494365460
// MI455X (gfx1250) — compile-verified
//
#include <hip/hip_runtime.h>
#include <hip/hip_bf16.h>

typedef __attribute__((ext_vector_type(2))) float v2f;
typedef __attribute__((ext_vector_type(8))) float v8f;

#define Bn   256
#define Cn   9
#define FRn  64
#define Tn   64
#define Pn   4
#define En   192
#define NFn  16
#define NTn  16
#define NTILES 12         // En / 16
#define NPATCH (Cn * NFn * NTn)   // 2304

// One wave (32 lanes) per (b, c, f) tile:
//   A tile: 16 t-patches x 16 (u*4+v) values  -> 4x V_WMMA_F32_16X16X4_F32 (K=16)
//   loops over 12 N-tiles of W (E = 192)
//   accumulator pre-loaded with bias + channel + spatial + freq + time embeddings
__global__ __launch_bounds__(32)
void patch_embed_wmma_f32(const float* __restrict__ x,
                          const float* __restrict__ W,       // (E, 16) row-major over k = 4u+v
                          const float* __restrict__ bias,    // (E)
                          const float* __restrict__ ch_emb,  // (C, E)
                          const float* __restrict__ sp_emb,  // (64, E)
                          const float* __restrict__ tpos,    // (NT, E)
                          const float* __restrict__ fpos,    // (NF, E)
                          const int*   __restrict__ sidx,    // (C)
                          float* __restrict__ out)           // (B, 2304, E)
{
    const int wave = blockIdx.x;          // 0 .. B*C*NF-1
    const int f  = wave % NFn;
    const int bc = wave / NFn;
    const int c  = bc % Cn;
    const int b  = bc / Cn;

    const int lane = threadIdx.x;         // 0..31 (wave32)
    const int lo   = lane & 15;           // column n (B/C/D) or row M (A)
    const int hi   = lane >> 4;           // half-wave select

    // ---- B tiles, resident in registers for the whole wave ----
    // B element (K-slice s, vgpr j, lane): W[(nt*16+lo)*16 + 4s + 2*hi + j]
    v2f Bt[NTILES][4];
    #pragma unroll
    for (int nt = 0; nt < NTILES; ++nt) {
        const float* wp = W + (nt * 16 + lo) * 16 + 2 * hi;
        #pragma unroll
        for (int s = 0; s < 4; ++s)
            Bt[nt][s] = *(const v2f*)(wp + 4 * s);
    }

    // ---- per-column additive term: bias + channel + spatial + freq ----
    const int sp = sidx[c];
    float base[NTILES];
    #pragma unroll
    for (int nt = 0; nt < NTILES; ++nt) {
        const int e = nt * 16 + lo;
        base[nt] = bias[e] + ch_emb[c * En + e] + sp_emb[sp * En + e]
                 + fpos[f * En + e];
    }

    // ---- A tile: x rows 4f..4f+3, each 64 floats, read exactly once ----
    // A element (K-slice s, vgpr j, lane): x[b,c,4f+s, 4*lo + 2*hi + j]
    v2f A[4];
    const float* xrow = x + (((size_t)b * Cn + c) * FRn + 4 * f) * Tn
                          + 4 * lo + 2 * hi;
    #pragma unroll
    for (int s = 0; s < 4; ++s)
        A[s] = __builtin_nontemporal_load((const v2f*)(xrow + s * Tn));

    // ---- output base for this (b,c,f) tile ----
    const int patch0 = (c * NFn + f) * NTn;            // + t
    float* outp = out + ((size_t)b * NPATCH + patch0) * En;

    #pragma unroll
    for (int nt = 0; nt < NTILES; ++nt) {
        const int e = nt * 16 + lo;

        // C init = base + time term; C/D vgpr g holds (t = g + 8*hi, n = lo)
        v8f acc;
        #pragma unroll
        for (int g = 0; g < 8; ++g)
            acc[g] = base[nt] + tpos[(g + 8 * hi) * En + e];

        // K = 16 as four accumulating 16x16x4 f32 WMMAs
        #pragma unroll
        for (int s = 0; s < 4; ++s)
            acc = __builtin_amdgcn_wmma_f32_16x16x4_f32(
                      false, A[s], false, Bt[nt][s],
                      (short)0, acc, false, false);

        // 8 coalesced stores (two 64B row segments per instruction), streamed NT
        #pragma unroll
        for (int g = 0; g < 8; ++g)
            __builtin_nontemporal_store(acc[g],
                outp + (size_t)(g + 8 * hi) * En + e);
    }
}

extern "C" void kernel_launch(void* const* d_in, const int* in_sizes, int n_in,
                              void* d_out, int out_size, void* d_ws, size_t ws_size,
                              hipStream_t stream) {
    const float* x      = (const float*)d_in[0];
    const float* W      = (const float*)d_in[1];
    const float* bias   = (const float*)d_in[2];
    const float* ch_emb = (const float*)d_in[3];
    const float* sp_emb = (const float*)d_in[4];
    const float* tpos   = (const float*)d_in[5];
    const float* fpos   = (const float*)d_in[6];
    const int*   sidx   = (const int*)d_in[7];
    float* out = (float*)d_out;

    const int nwaves = Bn * Cn * NFn;   // 36,864 single-wave workgroups
    patch_embed_wmma_f32<<<dim3(nwaves), dim3(32), 0, stream>>>(
        x, W, bias, ch_emb, sp_emb, tpos, fpos, sidx, out);
}